// Retriever_42339787604840
// MI455X (gfx1250) — compile-verified
//
#include <hip/hip_runtime.h>
#include <stdint.h>

// ---------------- problem constants ----------------
#define N_DOCS 500000
#define D_KEY  128
#define D_VAL  768
#define BS     256
#define TOPK   32

// ---------------- tiling constants -----------------
#define QCHUNK 64                       // queries per block (grid.y = BS/QCHUNK = 4)
#define DSUB   32                       // docs per inner iteration (2 WMMA tiles wide)
#define NDB    64                       // doc blocks (grid.x)
#define DOCS_PER_BLOCK ((N_DOCS + NDB - 1) / NDB)   // 7813
#define NCAND  (NDB * TOPK)             // 2048 candidates per query

// padded LDS strides (floats) to avoid bank conflicts
#define QSTR 132
#define KSTR 132
#define SSTR 36

typedef __attribute__((ext_vector_type(2))) float v2f;
typedef __attribute__((ext_vector_type(8))) float v8f;

#define NEG_INF (-__builtin_inff())

// =====================================================================
// Kernel 1: fused Q·K^T (f32 WMMA) + per-(query, doc-block) top-32.
// Key tiles are streamed with double-buffered GLOBAL_LOAD_ASYNC_TO_LDS_B128
// (ASYNCcnt-tracked, no VGPR staging), overlapping HBM with WMMA.
// =====================================================================
__global__ __launch_bounds__(256) void retr_score_topk_kernel(
    const float* __restrict__ Q,        // [BS, D_KEY]
    const float* __restrict__ Kd,       // [N_DOCS, D_KEY]
    float* __restrict__ cand_s,         // [BS, NDB, TOPK]
    int*   __restrict__ cand_i)         // [BS, NDB, TOPK]
{
    __shared__ float q_lds[QCHUNK * QSTR];      // 33.8 KB
    __shared__ float k_lds[2][DSUB * KSTR];     // 2 x 16.9 KB (double buffer)
    __shared__ float s_lds[QCHUNK * SSTR];      //  9.2 KB

    const int tid   = threadIdx.x;
    const int db    = blockIdx.x;               // doc block
    const int qb    = blockIdx.y;               // query chunk
    const int qbase = qb * QCHUNK;

    const long doc0 = (long)db * DOCS_PER_BLOCK;
    long doc1 = doc0 + DOCS_PER_BLOCK;
    if (doc1 > (long)N_DOCS) doc1 = (long)N_DOCS;
    const int T = (int)((doc1 - doc0 + DSUB - 1) / DSUB);

    // ---- stage the query chunk once (coalesced) ----
    for (int i = tid; i < QCHUNK * D_KEY; i += 256) {
        int r = i >> 7, c = i & 127;
        q_lds[r * QSTR + c] = Q[(long)(qbase + r) * D_KEY + c];
    }

    // async copy of one 32x128 f32 key tile into LDS buffer b:
    // 4 x b128 per thread (256 thr x 4 x 16B = 16 KB). OOB rows clamp to the
    // last doc row; their score columns are never read by the selector.
    auto issue_tile = [&](int tt, int b) {
        long dbase = doc0 + (long)tt * DSUB;
#pragma unroll
        for (int i = 0; i < 4; i++) {
            int e = i * 1024 + tid * 4;         // float index within tile
            int r = e >> 7, c = e & 127;        // c is a multiple of 4
            long d = dbase + r;
            if (d > (long)(N_DOCS - 1)) d = (long)(N_DOCS - 1);
            const float* src = Kd + d * D_KEY + c;
            unsigned lds_addr =
                (unsigned)(uintptr_t)&k_lds[b][r * KSTR + c]; // low 32b = LDS offset
            asm volatile("global_load_async_to_lds_b128 %0, %1, off"
                         :: "v"(lds_addr), "v"(src) : "memory");
        }
    };

    // ---- per-selector top-k state (threads 0..63 own one query each) ----
    float ts[TOPK]; int ti[TOPK];
    float tmin = NEG_INF; int tminpos = 0;
#pragma unroll
    for (int j = 0; j < TOPK; j++) { ts[j] = NEG_INF; ti[j] = -1; }

    // WMMA lane mapping (wave32)
    const int lane  = tid & 31;
    const int wave  = tid >> 5;              // 0..7
    const int qt    = wave >> 1;             // query tile 0..3 (16 rows each)
    const int dt    = wave & 1;              // doc tile 0..1  (16 cols each)
    const int mrow  = lane & 15;             // M (A) / N (B,C)
    const int khalf = lane >> 4;             // selects K pair {0,1} vs {2,3}

    issue_tile(0, 0);                        // prologue: tile 0 -> buf 0

    for (int t = 0; t < T; t++) {
        const int cur = t & 1, nxt = cur ^ 1;

        if (t + 1 < T) {
            issue_tile(t + 1, nxt);          // overlap next tile with compute
            if (t + 2 < T)                   // warm L2 two tiles ahead
                __builtin_prefetch(
                    Kd + (doc0 + (long)(t + 2) * DSUB) * D_KEY + (long)tid * 16,
                    0, 1);
            asm volatile("s_wait_asynccnt 0x4" ::: "memory"); // tile t landed
        } else {
            asm volatile("s_wait_asynccnt 0x0" ::: "memory");
        }
        __syncthreads();                     // all waves' tile-t writes visible
                                             // (also fences s_lds vs. selection)

        // ---- 16x16 score tile per wave via chained f32 WMMA over K=128 ----
        v8f acc = {};
        const float* aq = &q_lds[(qt * 16 + mrow) * QSTR + 2 * khalf];
        const float* bk = &k_lds[cur][(dt * 16 + mrow) * KSTR + 2 * khalf];
#pragma unroll
        for (int kk = 0; kk < D_KEY; kk += 4) {
            v2f a = *(const v2f*)(aq + kk);
            v2f b = *(const v2f*)(bk + kk);
            acc = __builtin_amdgcn_wmma_f32_16x16x4_f32(
                false, a, false, b, (short)0, acc, false, false);
        }
        // scatter C: row m = v + 8*khalf, col n = mrow
#pragma unroll
        for (int v = 0; v < 8; v++) {
            s_lds[(qt * 16 + v + 8 * khalf) * SSTR + dt * 16 + mrow] = acc[v];
        }
        __syncthreads();

        // ---- selection: thread t < 64 scans its query's 32 new scores ----
        if (tid < QCHUNK) {
            long dstart = doc0 + (long)t * DSUB;
            for (int d = 0; d < DSUB; d++) {
                long gd = dstart + d;
                if (gd >= doc1) break;
                float sc = s_lds[tid * SSTR + d];
                if (sc > tmin) {
                    ts[tminpos] = sc; ti[tminpos] = (int)gd;
                    tmin = ts[0]; tminpos = 0;
                    for (int j = 1; j < TOPK; j++)
                        if (ts[j] < tmin) { tmin = ts[j]; tminpos = j; }
                }
            }
        }
        // no trailing barrier needed: next iteration's post-wait barrier
        // orders s_lds writes after this selection, and the async writes of
        // tile t+1 target the buffer whose readers passed the mid barrier.
    }
    __syncthreads();

    // ---- emit candidate lists ----
    if (tid < QCHUNK) {
        long base = ((long)(qbase + tid) * NDB + db) * TOPK;
        for (int j = 0; j < TOPK; j++) {
            cand_s[base + j] = ts[j];
            cand_i[base + j] = ti[j];
        }
    }
}

// =====================================================================
// Kernel 2: per-query merge of 2048 candidates -> global top-32, gather
// =====================================================================
__global__ __launch_bounds__(256) void retr_merge_gather_kernel(
    const float* __restrict__ cand_s,
    const int*   __restrict__ cand_i,
    const float* __restrict__ Kd,       // [N_DOCS, D_KEY]
    const float* __restrict__ Vd,       // [N_DOCS, D_VAL]
    float* __restrict__ out_key,        // [BS, TOPK, D_KEY]
    float* __restrict__ out_val,        // [BS, TOPK, D_VAL]
    int*   __restrict__ out_I)          // [BS, TOPK]
{
    __shared__ float s_lds[NCAND];
    __shared__ int   i_lds[NCAND];
    __shared__ float red_s[256];
    __shared__ int   red_slot[256];
    __shared__ int   sel[TOPK];

    const int tid = threadIdx.x;
    const int q   = blockIdx.x;

    for (int i = tid; i < NCAND; i += 256) {
        s_lds[i] = cand_s[(long)q * NCAND + i];
        i_lds[i] = cand_i[(long)q * NCAND + i];
    }
    __syncthreads();

    // 32 rounds of block-wide argmax (ties -> lower doc index, like top_k)
    for (int j = 0; j < TOPK; j++) {
        float bs = NEG_INF; int bslot = -1; unsigned bdoc = 0xFFFFFFFFu;
        for (int i = tid; i < NCAND; i += 256) {
            float s = s_lds[i];
            unsigned d = (unsigned)i_lds[i];
            if (s > bs || (s == bs && d < bdoc)) { bs = s; bslot = i; bdoc = d; }
        }
        red_s[tid] = bs; red_slot[tid] = bslot;
        __syncthreads();
        for (int stride = 128; stride > 0; stride >>= 1) {
            if (tid < stride) {
                float s1 = red_s[tid],          s2 = red_s[tid + stride];
                int  sl1 = red_slot[tid],      sl2 = red_slot[tid + stride];
                unsigned d1 = (sl1 >= 0) ? (unsigned)i_lds[sl1] : 0xFFFFFFFFu;
                unsigned d2 = (sl2 >= 0) ? (unsigned)i_lds[sl2] : 0xFFFFFFFFu;
                if (s2 > s1 || (s2 == s1 && d2 < d1)) {
                    red_s[tid] = s2; red_slot[tid] = sl2;
                }
            }
            __syncthreads();
        }
        if (tid == 0) {
            int slot = red_slot[0];
            sel[j] = (slot >= 0) ? i_lds[slot] : 0;
            if (slot >= 0) s_lds[slot] = NEG_INF;   // remove winner
        }
        __syncthreads();
    }

    // ---- gather selected key rows ----
    for (int t = tid; t < TOPK * D_KEY; t += 256) {
        int j = t >> 7, c = t & 127;
        out_key[((long)q * TOPK + j) * D_KEY + c] = Kd[(long)sel[j] * D_KEY + c];
    }
    // ---- gather selected value rows ----
    for (int t = tid; t < TOPK * D_VAL; t += 256) {
        int j = t / D_VAL, c = t % D_VAL;
        out_val[((long)q * TOPK + j) * D_VAL + c] = Vd[(long)sel[j] * D_VAL + c];
    }
    // ---- indices ----
    if (tid < TOPK) out_I[q * TOPK + tid] = sel[tid];
}

// =====================================================================
extern "C" void kernel_launch(void* const* d_in, const int* in_sizes, int n_in,
                              void* d_out, int out_size, void* d_ws, size_t ws_size,
                              hipStream_t stream) {
    const float* Q  = (const float*)d_in[0];
    const float* Kd = (const float*)d_in[1];
    const float* Vd = (const float*)d_in[2];
    // d_in[3] is k (==32), baked in at compile time

    float* out     = (float*)d_out;
    float* out_key = out;
    float* out_val = out + (long)BS * TOPK * D_KEY;
    int*   out_I   = (int*)(out + (long)BS * TOPK * D_KEY + (long)BS * TOPK * D_VAL);

    float* cand_s = (float*)d_ws;                                   // 2 MB
    int*   cand_i = (int*)((char*)d_ws +
                           sizeof(float) * (size_t)BS * NDB * TOPK); // +2 MB

    dim3 gridA(NDB, BS / QCHUNK);
    retr_score_topk_kernel<<<gridA, 256, 0, stream>>>(Q, Kd, cand_s, cand_i);
    retr_merge_gather_kernel<<<BS, 256, 0, stream>>>(cand_s, cand_i, Kd, Vd,
                                                     out_key, out_val, out_I);
}